// Retriever_33062658245265
// MI455X (gfx1250) — compile-verified
//
#include <hip/hip_runtime.h>
#include <hip/hip_bf16.h>
#include <math.h>

typedef __attribute__((ext_vector_type(2))) float v2f;
typedef __attribute__((ext_vector_type(4))) float v4f;
typedef __attribute__((ext_vector_type(8))) float v8f;
typedef __attribute__((ext_vector_type(4))) int   v4i;

#define AS1 __attribute__((address_space(1)))
#define AS3 __attribute__((address_space(3)))

#define NQ 64
#define DIM 768
#define NTOT 100000
#define TOPK_K 5
#define EPS_SIM 1e-8f

#define TILE_N 128                 // K-rows per block
#define CHUNK 96                   // D-columns staged per LDS chunk (768 = 8*96)
#define NCHUNK (DIM / CHUNK)       // 8
#define KT_STRIDE 100              // ktile row stride in dwords (pad 96 -> 100, bank-conflict free)
#define SIMS_STRIDE 132            // sims row stride in dwords (pad 128 -> 132)
#define NBLOCKS ((NTOT + TILE_N - 1) / TILE_N)   // 782

// ---------------- CDNA5 async global->LDS staging (ASYNCcnt path) ----------------
__device__ __forceinline__ void async_copy_b128(const float* g, float* l) {
#if __has_builtin(__builtin_amdgcn_global_load_async_to_lds_b128)
    __builtin_amdgcn_global_load_async_to_lds_b128(
        (AS1 v4i*)g, (AS3 v4i*)l, 0, 0);
#else
    const unsigned lofs = (unsigned)(size_t)l;   // LDS byte offset = low 32 bits of generic addr
    asm volatile("global_load_async_to_lds_b128 %0, %1, off"
                 :: "v"(lofs), "v"(g) : "memory");
#endif
}

template <int N>
__device__ __forceinline__ void wait_asynccnt() {
#if __has_builtin(__builtin_amdgcn_s_wait_asynccnt)
    __builtin_amdgcn_s_wait_asynccnt(N);
#else
    asm volatile("s_wait_asynccnt %0" :: "i"(N) : "memory");
#endif
}

// issue the 12 async b128 copies this thread owns for one 128x96 chunk
__device__ __forceinline__ void stage_async_chunk(const float* gsrc_chunk, float* lds_base) {
#pragma unroll
    for (int j = 0; j < 12; ++j) {
        async_copy_b128(gsrc_chunk + j * 4, lds_base + j * 4);
    }
}

// -------- top-k helpers (tie-break: higher value, then lower index, like lax.top_k) ----------
__device__ __forceinline__ bool pair_better(float va, int ia, float vb, int ib) {
    return (va > vb) || ((va == vb) && ((unsigned)ia < (unsigned)ib));
}

__device__ __forceinline__ void top5_insert(float v, int i, float* bv, int* bi) {
#pragma unroll
    for (int k = 0; k < TOPK_K; ++k) {
        if (pair_better(v, i, bv[k], bi[k])) {
#pragma unroll
            for (int s = TOPK_K - 1; s > k; --s) { bv[s] = bv[s - 1]; bi[s] = bi[s - 1]; }
            bv[k] = v; bi[k] = i;
            break;
        }
    }
}

// -------------------- kernel 0: query norms --------------------
__global__ __launch_bounds__(256) void qnorm_kernel(const float* __restrict__ q,
                                                    float* __restrict__ qnorm) {
    __shared__ float red[256];
    const int b = blockIdx.x;    // query
    const int t = threadIdx.x;
    float s = 0.f;
    for (int i = t; i < DIM; i += 256) { const float v = q[(size_t)b * DIM + i]; s += v * v; }
    red[t] = s;
    __syncthreads();
    for (int st = 128; st > 0; st >>= 1) {
        if (t < st) red[t] += red[t + st];
        __syncthreads();
    }
    if (t == 0) qnorm[b] = sqrtf(red[0]);
}

// -- kernel 1: double-buffered async-LDS staging + WMMA fp32 GEMM (64x128) + fused norms + top-5 --
__global__ __launch_bounds__(256) void sim_topk_kernel(
    const float* __restrict__ Q, const float* __restrict__ KE,
    const float* __restrict__ qnorm,
    float* __restrict__ candVal, int* __restrict__ candIdx)
{
    __shared__ float ktile[2][TILE_N * KT_STRIDE];   // 2 x 51200 B (double buffer)
    __shared__ float sims[NQ * SIMS_STRIDE];         // 33792 B
    __shared__ float nrmPart[256];
    __shared__ float knorm[TILE_N];

    const int t = threadIdx.x;
    const int nbase = blockIdx.x * TILE_N;

    // WMMA fragment indexing (wave32)
    const int l = t & 31;
    const int w = t >> 5;                 // wave 0..7
    const int mtile = w & 3;              // query M-tile 0..3 (16 rows each)
    const int nsBase = (w >> 2) * 4;      // N-subtile group base: 0 or 4
    const int lm = l & 15;                // row/col within 16
    const int khalf = (l >> 4) << 1;      // K offset per lane half: 0 or 2
    const int mrow = mtile * 16 + lm;     // this lane's query row for A frag

    // staging: 2 threads per K-row, 48 floats (12 x b128) each per chunk
    const int lrow = t >> 1;
    const int lcol = (t & 1) * 48;
    const bool inRange = (nbase + lrow) < NTOT;
    const float* gsrc = KE + (size_t)(nbase + lrow) * DIM + lcol;
    const int lbase = lrow * KT_STRIDE + lcol;

    // prologue: async-stage chunk 0 into buffer 0
    if (inRange) stage_async_chunk(gsrc, &ktile[0][lbase]);

    v8f acc[4] = {};      // 4 N-subtiles of 16x16 f32 accumulators
    float sq = 0.f;       // partial ||k||^2 for (lrow, lcol half)

    for (int chunk = 0; chunk < NCHUNK; ++chunk) {
        const int buf = chunk & 1;
        // async-stage next chunk into the other buffer, then wait for current chunk's 12 copies
        if (chunk + 1 < NCHUNK) {
            if (inRange) stage_async_chunk(gsrc + (chunk + 1) * CHUNK, &ktile[buf ^ 1][lbase]);
            wait_asynccnt<12>();          // in-order completion: current chunk has landed
        } else {
            wait_asynccnt<0>();
        }
        __syncthreads();

        // ---- WMMA over this chunk: K stepped by 4 ----
        const int c0 = chunk * CHUNK;
        const float* arow = Q + (size_t)mrow * DIM + c0 + khalf;
        const float* ktb = &ktile[buf][0];
        for (int kk = 0; kk < CHUNK; kk += 4) {
            const v2f a = *reinterpret_cast<const v2f*>(arow + kk);   // A: M=lm, K=kk+khalf+{0,1}
#pragma unroll
            for (int j = 0; j < 4; ++j) {
                const int nloc = (nsBase + j) * 16 + lm;              // B: N=lm, K=kk+khalf+{0,1}
                const v2f b = *reinterpret_cast<const v2f*>(ktb + nloc * KT_STRIDE + kk + khalf);
                acc[j] = __builtin_amdgcn_wmma_f32_16x16x4_f32(
                    false, a, false, b, (short)0, acc[j], false, false);
            }
        }

        // ---- accumulate ||k||^2 from the staged tile (read-back) ----
#pragma unroll
        for (int j = 0; j < 12; ++j) {
            const v4f v = *reinterpret_cast<const v4f*>(ktb + lbase + j * 4);
            sq += v.x * v.x + v.y * v.y + v.z * v.z + v.w * v.w;
        }
        __syncthreads();
    }

    // ---- finalize k-norms for this tile ----
    nrmPart[t] = sq;
    __syncthreads();
    if (t < TILE_N) knorm[t] = sqrtf(nrmPart[2 * t] + nrmPart[2 * t + 1]);

    // ---- spill accumulators to sims[64][128] (C layout: vgpr r -> M = r + 8*(l/16)) ----
#pragma unroll
    for (int j = 0; j < 4; ++j) {
#pragma unroll
        for (int r = 0; r < 8; ++r) {
            const int m = mtile * 16 + r + ((l >> 4) << 3);
            const int n = (nsBase + j) * 16 + lm;
            sims[m * SIMS_STRIDE + n] = acc[j][r];
        }
    }
    __syncthreads();

    // ---- per-query top-5 within this 128-column tile ----
    if (t < NQ) {
        const float qn = qnorm[t];
        float bv[TOPK_K]; int bi[TOPK_K];
#pragma unroll
        for (int k = 0; k < TOPK_K; ++k) { bv[k] = -INFINITY; bi[k] = -1; }
        for (int n = 0; n < TILE_N; ++n) {
            const int g = nbase + n;
            if (g >= NTOT) break;          // rows >= N carry stale LDS; never considered
            const float val = sims[t * SIMS_STRIDE + n] / fmaxf(qn * knorm[n], EPS_SIM);
            top5_insert(val, g, bv, bi);
        }
        const size_t base = ((size_t)blockIdx.x * NQ + t) * TOPK_K;
#pragma unroll
        for (int k = 0; k < TOPK_K; ++k) { candVal[base + k] = bv[k]; candIdx[base + k] = bi[k]; }
    }
}

// -------------------- kernel 2: merge per-block candidates -> global top-5 per query ----------
__global__ __launch_bounds__(256) void topk_merge_kernel(
    const float* __restrict__ candVal, const int* __restrict__ candIdx,
    int* __restrict__ topIdx)
{
    __shared__ float lv[256 * TOPK_K];
    __shared__ int   li[256 * TOPK_K];
    const int q = blockIdx.x;
    const int t = threadIdx.x;
    const int M = NBLOCKS * TOPK_K;   // 3910 candidates for this query

    float bv[TOPK_K]; int bi[TOPK_K];
#pragma unroll
    for (int k = 0; k < TOPK_K; ++k) { bv[k] = -INFINITY; bi[k] = -1; }
    for (int c = t; c < M; c += 256) {
        const int b = c / TOPK_K;
        const int k = c - b * TOPK_K;
        const size_t off = ((size_t)b * NQ + q) * TOPK_K + k;
        top5_insert(candVal[off], candIdx[off], bv, bi);
    }
#pragma unroll
    for (int k = 0; k < TOPK_K; ++k) { lv[t * TOPK_K + k] = bv[k]; li[t * TOPK_K + k] = bi[k]; }
    __syncthreads();

    // merge tree of sorted-descending 5-lists
    for (int s = 128; s >= 1; s >>= 1) {
        if (t < s) {
            float av[TOPK_K], cv[TOPK_K]; int ai[TOPK_K], ci[TOPK_K];
#pragma unroll
            for (int k = 0; k < TOPK_K; ++k) {
                av[k] = lv[t * TOPK_K + k];        ai[k] = li[t * TOPK_K + k];
                cv[k] = lv[(t + s) * TOPK_K + k];  ci[k] = li[(t + s) * TOPK_K + k];
            }
            float ov[TOPK_K]; int oi[TOPK_K];
            int pa = 0, pb = 0;
#pragma unroll
            for (int k = 0; k < TOPK_K; ++k) {
                const bool ta = pair_better(av[pa], ai[pa], cv[pb], ci[pb]);
                ov[k] = ta ? av[pa] : cv[pb];
                oi[k] = ta ? ai[pa] : ci[pb];
                if (ta) ++pa; else ++pb;
            }
#pragma unroll
            for (int k = 0; k < TOPK_K; ++k) { lv[t * TOPK_K + k] = ov[k]; li[t * TOPK_K + k] = oi[k]; }
        }
        __syncthreads();
    }
    if (t == 0) {
#pragma unroll
        for (int k = 0; k < TOPK_K; ++k) topIdx[q * TOPK_K + k] = li[k];
    }
}

// -------------------- kernel 3: gather outputs --------------------
// d_out layout (flat, return order): topk_knowledge [64,5,32] as float, then topk_embed [64,5,768]
__global__ __launch_bounds__(256) void gather_kernel(
    const float* __restrict__ KE, const int* __restrict__ KF,
    const int* __restrict__ topIdx, float* __restrict__ out)
{
    const int p = blockIdx.x;    // 0..319 = q*5 + k
    const int t = threadIdx.x;
    const int idx = topIdx[p];
    if (t < 32) out[(size_t)p * 32 + t] = (float)KF[(size_t)idx * 32 + t];
    const float* src = KE + (size_t)idx * DIM;
    float* dst = out + (size_t)NQ * TOPK_K * 32 + (size_t)p * DIM;
    for (int i = t; i < DIM; i += 256) dst[i] = src[i];
}

// -------------------- host entry --------------------
extern "C" void kernel_launch(void* const* d_in, const int* in_sizes, int n_in,
                              void* d_out, int out_size, void* d_ws, size_t ws_size,
                              hipStream_t stream)
{
    const float* Q  = (const float*)d_in[0];   // [64, 768] f32
    const float* KE = (const float*)d_in[1];   // [100000, 768] f32
    const int*   KF = (const int*)d_in[2];     // [100000, 32] i32
    float* out = (float*)d_out;
    char* ws = (char*)d_ws;

    // workspace layout (~2 MB total)
    float* qnorm = (float*)ws;                                        // 64 f32
    float* candVal = (float*)(ws + 1024);                             // NBLOCKS*64*5 f32
    const size_t candCount = (size_t)NBLOCKS * NQ * TOPK_K;           // 250240
    int* candIdx = (int*)(ws + 1024 + candCount * sizeof(float));     // NBLOCKS*64*5 i32
    int* topIdx  = (int*)(ws + 1024 + candCount * (sizeof(float) + sizeof(int)));  // 320 i32

    qnorm_kernel<<<NQ, 256, 0, stream>>>(Q, qnorm);
    sim_topk_kernel<<<NBLOCKS, 256, 0, stream>>>(Q, KE, qnorm, candVal, candIdx);
    topk_merge_kernel<<<NQ, 256, 0, stream>>>(candVal, candIdx, topIdx);
    gather_kernel<<<NQ * TOPK_K, 256, 0, stream>>>(KE, KF, topIdx, out);
}